// GeneralizedRecallloss_3633542333177
// MI455X (gfx1250) — compile-verified
//
#include <hip/hip_runtime.h>
#include <math.h>

// Problem geometry fixed by the reference: scores (B x 8192), P=16 positives,
// k_vals length K (=3). Output is one float scalar.
#define ROWLEN 8192
#define PNUM   16
#define BLOCK  256
#define NEGN   (ROWLEN - PNUM)                 // 8176
#define NCHUNK ((NEGN + BLOCK - 1) / BLOCK)    // 32

// CDNA5 async copy: global -> LDS, tracked by ASYNCcnt (CDNA5 ISA).
// GVS addressing: mem = SADDR(64b sgpr pair) + VADDR(32b per-lane byte offset).
// First operand is the LDS byte address (wave-relative; low 32 bits of a generic
// __shared__ pointer are exactly that per the flat-address aperture rules).
__device__ __forceinline__ void async_copy_f32(unsigned lds_addr, unsigned byte_off,
                                               const float* base) {
  asm volatile("global_load_async_to_lds_b32 %0, %1, %2"
               :: "v"(lds_addr), "v"(byte_off), "s"(base)
               : "memory");
}

__global__ __launch_bounds__(BLOCK)
void recall_row_kernel(const float* __restrict__ scores,
                       const int* __restrict__ kvals, int K,
                       float* __restrict__ partials) {
  __shared__ float spos[PNUM];
  __shared__ float sbuf[2][BLOCK];
  __shared__ float wred[BLOCK / 32][PNUM];
  __shared__ float jred[PNUM];

  const int b = blockIdx.x;
  const int t = threadIdx.x;
  const float* row = scores + (size_t)b * ROWLEN;

  if (t < PNUM) spos[t] = row[t];
  __syncthreads();

  float pjm[PNUM];   // pos[j] - NEG_MARGIN, so d = neg - pos + margin
  float acc[PNUM];
#pragma unroll
  for (int j = 0; j < PNUM; ++j) {
    pjm[j] = spos[j] - 0.1f;
    acc[j] = 0.0f;
  }

  const unsigned lds0 = (unsigned)(unsigned long long)(void*)&sbuf[0][t];
  const unsigned lds1 = (unsigned)(unsigned long long)(void*)&sbuf[1][t];

  // Prologue: stage chunk 0 (always full: NEGN >= BLOCK).
  async_copy_f32(lds0, (unsigned)((PNUM + t) * 4), row);

  for (int c = 0; c < NCHUNK; ++c) {
    if (c + 1 < NCHUNK) {
      // Prefetch chunk c+1 into the other buffer; clamp tail addresses so the
      // instruction always issues with full EXEC (lanes past the end just load
      // an in-row byte they never read).
      unsigned nn = (unsigned)(PNUM + (c + 1) * BLOCK + t);
      if (nn >= ROWLEN) nn = ROWLEN - 1;
      async_copy_f32((c & 1) ? lds0 : lds1, nn * 4u, row);
      asm volatile("s_wait_asynccnt 1" ::: "memory");   // chunk c landed
    } else {
      asm volatile("s_wait_asynccnt 0" ::: "memory");   // drain last chunk
    }

    const int n = PNUM + c * BLOCK + t;
    if (n < ROWLEN) {
      const float v = sbuf[c & 1][t];
#pragma unroll
      for (int j = 0; j < PNUM; ++j) {
        const float d   = v - pjm[j];                // neg - pos + margin
        const bool  ge  = (d >= 0.0f);
        // Select the temperature first so only ONE correctly-rounded IEEE
        // divide is issued per pair (matches reference d/0.1 resp. d/0.01).
        const float tmp = ge ? 0.1f : 0.01f;
        const float x   = d / tmp;
        // logaddexp(0,x) = max(x,0) + log1p(exp(-|x|))  (matches jnp.logaddexp)
        const float sp  = fmaxf(x, 0.0f) + log1pf(expf(-fabsf(x)));
        acc[j] += ge ? (sp + 0.1f) : sp;
      }
    }
  }

  // Wave32 butterfly reduction of the 16 accumulators, then cross-wave via LDS.
  const int lane = t & 31;
  const int wave = t >> 5;
#pragma unroll
  for (int j = 0; j < PNUM; ++j) {
#pragma unroll
    for (int m = 16; m >= 1; m >>= 1)
      acc[j] += __shfl_xor(acc[j], m, 32);
  }
  if (lane == 0) {
#pragma unroll
    for (int j = 0; j < PNUM; ++j) wred[wave][j] = acc[j];
  }
  __syncthreads();

  if (t < PNUM) {
    const int j = t;
    float negsum = 0.0f;
#pragma unroll
    for (int w = 0; w < BLOCK / 32; ++w) negsum += wred[w][j];

    // pos_term: count of i != j with pos[i] - pos[j] >= 0
    float poscnt = 0.0f;
    const float pj = spos[j];
#pragma unroll
    for (int i = 0; i < PNUM; ++i)
      if (i != j && (spos[i] - pj) >= 0.0f) poscnt += 1.0f;

    const float rank = 1.0f + poscnt + negsum;
    float rsum = 0.0f;
    for (int kk = 0; kk < K; ++kk) {
      float z = (rank - (float)kvals[kk]) / 0.1f;
      z = fminf(z, 88.0f);                               // CLAMP_MAX
      rsum += fmaxf(z, 0.0f) + log1pf(expf(-fabsf(z)));  // logaddexp(0, z)
    }
    jred[j] = rsum;
  }
  __syncthreads();

  if (t == 0) {
    float s = 0.0f;
#pragma unroll
    for (int j = 0; j < PNUM; ++j) s += jred[j];
    partials[b] = s / (float)(PNUM * K);
  }
}

// Deterministic fixed-order final mean (no float atomics -> bitwise repeatable).
__global__ __launch_bounds__(BLOCK)
void recall_reduce_kernel(const float* __restrict__ part, int B,
                          float* __restrict__ out) {
  __shared__ float s[BLOCK];
  const int t = threadIdx.x;
  float v = 0.0f;
  for (int i = t; i < B; i += BLOCK) v += part[i];
  s[t] = v;
  __syncthreads();
  for (int off = BLOCK / 2; off > 0; off >>= 1) {
    if (t < off) s[t] += s[t + off];
    __syncthreads();
  }
  if (t == 0) out[0] = s[0] / (float)B;
}

extern "C" void kernel_launch(void* const* d_in, const int* in_sizes, int n_in,
                              void* d_out, int out_size, void* d_ws, size_t ws_size,
                              hipStream_t stream) {
  const float* scores = (const float*)d_in[0];
  const int*   kvals  = (const int*)d_in[1];
  const int    K      = in_sizes[1];            // 3
  const int    B      = in_sizes[0] / ROWLEN;   // 512
  float* partials = (float*)d_ws;               // B floats of scratch

  recall_row_kernel<<<B, BLOCK, 0, stream>>>(scores, kvals, K, partials);
  recall_reduce_kernel<<<1, BLOCK, 0, stream>>>(partials, B, (float*)d_out);
}